// KVGather_12670153523749
// MI455X (gfx1250) — compile-verified
//
#include <hip/hip_runtime.h>

// ---------------------------------------------------------------------------
// KV gather + routing-weight scale (memory-bound streaming gather).
//   r_idx    : [N, P2, TOPK]  int   (region indices into P2)
//   r_weight : [N, P2, TOPK]  f32
//   kv       : [N, P2, W2, C_KV] f32   (region = W2*C_KV = 32768 f32 = 128 KB)
//   out      : [N, P2, TOPK, W2, C_KV] f32 = r_weight * kv[n, r_idx]
//
// Strategy (MI455X):
//  * One 256-thread block per (n,i,k) triple -> contiguous 128 KB copy,
//    scaled by one scalar weight (roofline: 0.25 FLOP/byte, pure bandwidth).
//  * CDNA5 async data mover: GLOBAL_LOAD_ASYNC_TO_LDS_B128, 4-deep pipeline.
//    Each lane's async copy lands in lane-private LDS words, so NO barriers
//    are needed: per-wave s_wait_asynccnt orders producer->consumer, and the
//    compiler's s_wait_dscnt (for the scale mul) orders consumer->reuse.
//  * Reads stay RT (kv = 128 MiB fits the 192 MB L2); output (1 GiB, never
//    re-read) streams out with non-temporal b128 stores to keep L2 clean.
// ---------------------------------------------------------------------------

typedef float v4f __attribute__((ext_vector_type(4)));
typedef int   v4i __attribute__((__vector_size__(16)));   // matches builtin proto

#define AS_GLOBAL __attribute__((address_space(1)))
#define AS_LOCAL  __attribute__((address_space(3)))

// ---- CDNA5 async global->LDS 16B copy (per lane) ---------------------------
__device__ __forceinline__ void async_g2l_b128(const float* g, float* l) {
#if defined(__has_builtin) && __has_builtin(__builtin_amdgcn_global_load_async_to_lds_b128)
  __builtin_amdgcn_global_load_async_to_lds_b128(
      (AS_GLOBAL v4i*)(g), (AS_LOCAL v4i*)(l), 0, 0);
#else
  unsigned loff = (unsigned)(__SIZE_TYPE__)(AS_LOCAL char*)(l);
  unsigned long long ga = (unsigned long long)(g);
  asm volatile("global_load_async_to_lds_b128 %0, %1, off"
               :: "v"(loff), "v"(ga)
               : "memory");
#endif
}

// ---- s_wait_asynccnt <= n (n must be a source-level constant) --------------
#if defined(__has_builtin) && __has_builtin(__builtin_amdgcn_s_wait_asynccnt)
#define WAIT_ASYNCCNT(n) __builtin_amdgcn_s_wait_asynccnt(n)
#else
#define WAIT_ASYNCCNT(n) asm volatile("s_wait_asynccnt %0" :: "i"(n) : "memory")
#endif

namespace {
constexpr int kP2      = 64;
constexpr int kW2      = 64;
constexpr int kCkv     = 512;
constexpr int kTopK    = 8;
constexpr int kRegion  = kW2 * kCkv;         // 32768 floats (128 KB) per region
constexpr int kThreads = 256;                // 8 wave32
constexpr int kStep    = kThreads * 4;       // 1024 floats (4 KB) per stage
constexpr int kNSteps  = kRegion / kStep;    // 32 stages per block
constexpr int kDepth   = 4;                  // async pipeline depth (16 KB LDS)
}

__global__ __launch_bounds__(kThreads)
void KVGather_12670153523749_kernel(const int*   __restrict__ r_idx,
                                    const float* __restrict__ r_weight,
                                    const float* __restrict__ kv,
                                    float*       __restrict__ out) {
  // kDepth-deep LDS staging ring: kDepth x 4 KB (lane-private slots).
  __shared__ v4f lds4[kDepth * kThreads];

  const int b = blockIdx.x;                  // (n*P2 + i)*TOPK + k, uniform
  const int t = threadIdx.x;
  const int n = b / (kP2 * kTopK);
  const int idx = r_idx[b];                  // region index in [0, P2)
  const float w = r_weight[b];

  const float* __restrict__ src = kv  + ((size_t)(n * kP2 + idx) * kRegion);
  float*       __restrict__ dst = out + ((size_t)b * kRegion);

  // Prologue: fill the pipeline (stages 0..kDepth-1).
#pragma unroll
  for (int d = 0; d < kDepth; ++d)
    async_g2l_b128(src + d * kStep + t * 4, (float*)&lds4[d * kThreads + t]);

#pragma unroll
  for (int s = 0; s < kNSteps; ++s) {
    // Steady state: kDepth stages outstanding -> wait <= kDepth-1 completes
    // the oldest (async loads complete in order). Tail: wait for all.
    if (s <= kNSteps - kDepth) {
      WAIT_ASYNCCNT(kDepth - 1);
    } else {
      WAIT_ASYNCCNT(0);
    }

    v4f v = lds4[(s % kDepth) * kThreads + t];   // ds_load_b128 (lane-private)
    v *= w;                                      // packed f32 scale
    // Streaming output: non-temporal b128 store, keep kv resident in L2.
    __builtin_nontemporal_store(v, (v4f*)(dst + (size_t)s * kStep + t * 4));

    // Refill the just-consumed slot (ds_load already retired: the mul above
    // forced s_wait_dscnt before we reach this issue in program order).
    if (s + kDepth < kNSteps)
      async_g2l_b128(src + (s + kDepth) * kStep + t * 4,
                     (float*)&lds4[(s % kDepth) * kThreads + t]);
  }
}

extern "C" void kernel_launch(void* const* d_in, const int* in_sizes, int n_in,
                              void* d_out, int out_size, void* d_ws, size_t ws_size,
                              hipStream_t stream) {
  const int*   r_idx    = (const int*)d_in[0];    // integer input -> const int*
  const float* r_weight = (const float*)d_in[1];
  const float* kv       = (const float*)d_in[2];
  float*       out      = (float*)d_out;

  const int n_triples = in_sizes[0];              // N*P2*TOPK = 8192 blocks
  KVGather_12670153523749_kernel<<<n_triples, kThreads, 0, stream>>>(
      r_idx, r_weight, kv, out);
}